// One_shot_double_loss_24773371363368
// MI455X (gfx1250) — compile-verified
//
#include <hip/hip_runtime.h>

typedef __attribute__((ext_vector_type(2))) float v2f;
typedef __attribute__((ext_vector_type(8))) float v8f;

#define N_PTS 8192
#define DIMS  64
#define TILE  16
#define NTILE (N_PTS / TILE)      // 512 tiles per dimension
#define WAVES_PER_BLOCK 8         // 256 threads, wave32

// Workspace layout (floats): [0..N) row sq-norms | [N..N+2) float sums | [N+2..N+4) u32 counts

// ---------------------------------------------------------------------------
// Kernel 1: per-row squared norms + zero the global accumulators.
// ---------------------------------------------------------------------------
__global__ __launch_bounds__(256)
void sqnorm_init_kernel(const float* __restrict__ P, float* __restrict__ sq,
                        float* __restrict__ facc, unsigned int* __restrict__ cacc) {
    const int idx = blockIdx.x * 256 + threadIdx.x;
    if (idx < 2) { facc[idx] = 0.0f; cacc[idx] = 0u; }
    if (idx < N_PTS) {
        const float4* row = (const float4*)(P + (size_t)idx * DIMS);
        float s = 0.0f;
        #pragma unroll
        for (int k = 0; k < DIMS / 4; ++k) {
            float4 v = row[k];
            s += v.x * v.x + v.y * v.y + v.z * v.z + v.w * v.w;
        }
        sq[idx] = s;
    }
}

// ---------------------------------------------------------------------------
// Kernel 2: one wave per 16x16 tile of the Gram matrix via f32 WMMA.
// ---------------------------------------------------------------------------
__global__ __launch_bounds__(256)
void pair_tile_kernel(const float* __restrict__ P, const int* __restrict__ tgt,
                      const float* __restrict__ sq,
                      float* __restrict__ facc, unsigned int* __restrict__ cacc) {
    __shared__ float        s_f[2];
    __shared__ unsigned int s_c[2];
    if (threadIdx.x == 0) { s_f[0] = 0.0f; s_f[1] = 0.0f; s_c[0] = 0u; s_c[1] = 0u; }
    __syncthreads();

    const int lane = threadIdx.x & 31;
    const int wave = threadIdx.x >> 5;
    const int ti   = blockIdx.y * WAVES_PER_BLOCK + wave;   // tile row
    const int tj   = blockIdx.x;                            // tile col
    const int i0   = ti * TILE;
    const int j0   = tj * TILE;

    const int half = lane >> 4;     // 0: lanes 0-15, 1: lanes 16-31
    const int m    = lane & 15;

    // ISA 16x4 f32 A layout: lane = m (row), VGPR0/1 <- K = 2*half + {0,1}.
    // B (4x16) mirrors it with N = m. Both tiles load rows of P identically.
    const float* pa = P + (size_t)(i0 + m) * DIMS + 2 * half;
    const float* pb = P + (size_t)(j0 + m) * DIMS + 2 * half;

    v8f c = {};   // C/D: VGPR r <-> (M = r + 8*half, N = m)
    #pragma unroll
    for (int kk = 0; kk < DIMS / 4; ++kk) {
        v2f a = *(const v2f*)(pa + kk * 4);
        v2f b = *(const v2f*)(pb + kk * 4);
        // D = A x B + C, exact f32 path: v_wmma_f32_16x16x4_f32
        c = __builtin_amdgcn_wmma_f32_16x16x4_f32(
                /*neg_a=*/false, a, /*neg_b=*/false, b,
                /*c_mod=*/(short)0, c, /*reuse_a=*/false, /*reuse_b=*/false);
    }

    float        same_sum = 0.0f, diff_sum = 0.0f;
    unsigned int scnt = 0u, dcnt = 0u;

    const int   j     = j0 + m;
    const float sqj   = sq[j];
    const int   clsj  = tgt[j];
    #pragma unroll
    for (int r = 0; r < 8; ++r) {
        const int i = i0 + r + 8 * half;
        // ti > tj  =>  i >= i0 > j0+15 >= j, so this mask also kills the
        // redundant lower-triangle tiles (which ran WMMA with full EXEC).
        if (i < j) {
            float d2 = sq[i] + sqj - 2.0f * c[r];
            d2 = fmaxf(d2, 0.0f);
            const float dist = sqrtf(d2);
            if (tgt[i] == clsj) { same_sum += dist;        ++scnt; }
            else                { diff_sum += 1.0f / dist; ++dcnt; }
        }
    }

    atomicAdd(&s_f[0], same_sum);
    atomicAdd(&s_f[1], diff_sum);
    atomicAdd(&s_c[0], scnt);
    atomicAdd(&s_c[1], dcnt);
    __syncthreads();
    if (threadIdx.x == 0) {
        atomicAdd(&facc[0], s_f[0]);
        atomicAdd(&facc[1], s_f[1]);
        atomicAdd(&cacc[0], s_c[0]);
        atomicAdd(&cacc[1], s_c[1]);
    }
}

// ---------------------------------------------------------------------------
// Kernel 3: finalize -> (loss, same_loss, diff_loss)   [ALPHA=BETA=WEIGHT=1]
// ---------------------------------------------------------------------------
__global__ void finalize_kernel(const float* __restrict__ facc,
                                const unsigned int* __restrict__ cacc,
                                float* __restrict__ out) {
    if (threadIdx.x == 0 && blockIdx.x == 0) {
        const float same_sum = facc[0];
        const float diff_sum = facc[1];
        const unsigned int sn = cacc[0];
        const unsigned int dn = cacc[1];
        const float same_loss = (sn > 0u) ? same_sum / (float)sn : same_sum;
        const float diff_loss = (dn > 0u) ? diff_sum / (float)dn : diff_sum;
        out[0] = same_loss + diff_loss;
        out[1] = same_loss;
        out[2] = diff_loss;
    }
}

extern "C" void kernel_launch(void* const* d_in, const int* in_sizes, int n_in,
                              void* d_out, int out_size, void* d_ws, size_t ws_size,
                              hipStream_t stream) {
    const float* P   = (const float*)d_in[0];   // [8192, 64] f32
    const int*   tgt = (const int*)d_in[1];     // [8192] i32

    float*        ws_f = (float*)d_ws;
    float*        sq   = ws_f;                           // N floats
    float*        facc = ws_f + N_PTS;                   // 2 floats
    unsigned int* cacc = (unsigned int*)(ws_f + N_PTS + 2); // 2 u32

    sqnorm_init_kernel<<<N_PTS / 256, 256, 0, stream>>>(P, sq, facc, cacc);

    dim3 grid(NTILE, NTILE / WAVES_PER_BLOCK);  // 512 x 64 blocks, 1 wave/tile
    pair_tile_kernel<<<grid, 256, 0, stream>>>(P, tgt, sq, facc, cacc);

    finalize_kernel<<<1, 32, 0, stream>>>(facc, cacc, (float*)d_out);
}